// PromptInjectionGraphGPS_52733608460568
// MI455X (gfx1250) — compile-verified
//
#include <hip/hip_runtime.h>
#include <math.h>

// ---------------- problem constants ----------------
#define NN    16384      // total nodes (B*NPG)
#define HID   256
#define BATCH 64
#define NPG   256
#define HEADS 8
#define HDIM  32
#define EDG   262144     // N*16
#define FIN   768
#define FFNH  1024

typedef __attribute__((ext_vector_type(2))) float v2f;
typedef __attribute__((ext_vector_type(8))) float v8f;

// D = A(16x4,f32) * B(4x16,f32) + C  -- full fp32 WMMA (CDNA5)
__device__ __forceinline__ v8f wmma4(v2f a, v2f b, v8f c) {
  return __builtin_amdgcn_wmma_f32_16x16x4_f32(false, a, false, b, (short)0, c,
                                               false, false);
}

// ---------------------------------------------------------------
// fp32 WMMA GEMM: C[M,Nd] = A[M,K] @ B[K,Nd] (+bias)(+relu)
// One wave computes a 16x64 strip (4 accumulators) so each A fragment
// feeds 4 WMMAs. Inner loop is pure pointer-bump + constant-offset loads.
// Requires: M % 16 == 0, Nd % 64 == 0, K % 4 == 0.
// A-frag: lane<16 -> A[m0+ml][k0+{0,1}], lane>=16 -> A[m0+ml][k0+{2,3}]
// B-frag: lane<16 -> B[k0+{0,1}][n+ml],  lane>=16 -> B[k0+{2,3}][n+ml]
// C/D   : VGPR r -> row m0+r (lanes 0-15) / m0+r+8 (lanes 16-31)
// ---------------------------------------------------------------
__global__ void gemm_wmma_f32(const float* __restrict__ A,
                              const float* __restrict__ B,
                              const float* __restrict__ bias,
                              float* __restrict__ C,
                              int M, int Nd, int K, int relu) {
  const int lane = threadIdx.x & 31;
  const int wave = blockIdx.x * (blockDim.x >> 5) + (threadIdx.x >> 5);
  const int nstrips = Nd >> 6;
  const int total   = (M >> 4) * nstrips;
  if (wave >= total) return;                 // wave-uniform: EXEC stays all-ones
  const int m0 = (wave / nstrips) << 4;
  const int n0 = (wave % nstrips) << 6;
  const int ml   = lane & 15;
  const int koff = (lane >> 4) << 1;         // 0 or 2

  v8f acc0 = {}, acc1 = {}, acc2 = {}, acc3 = {};
  const float* ap  = A + (size_t)(m0 + ml) * K + koff;       // 8B aligned
  const float* bp0 = B + (size_t)koff * Nd + n0 + ml;        // row k0+koff
  const float* bp1 = bp0 + Nd;                               // row k0+koff+1
  const size_t bstep = (size_t)4 * Nd;

  for (int k0 = 0; k0 < K; k0 += 4) {
    v2f a = *(const v2f*)ap;
    v2f b0, b1, b2, b3;
    b0.x = bp0[0];  b0.y = bp1[0];
    b1.x = bp0[16]; b1.y = bp1[16];
    b2.x = bp0[32]; b2.y = bp1[32];
    b3.x = bp0[48]; b3.y = bp1[48];
    acc0 = wmma4(a, b0, acc0);
    acc1 = wmma4(a, b1, acc1);
    acc2 = wmma4(a, b2, acc2);
    acc3 = wmma4(a, b3, acc3);
    ap  += 4;
    bp0 += bstep;
    bp1 += bstep;
  }

  const int rbase = m0 + ((lane >> 4) << 3);
  v8f accs[4] = {acc0, acc1, acc2, acc3};
  #pragma unroll
  for (int t = 0; t < 4; ++t) {
    const int col  = n0 + t * 16 + ml;
    const float bv = bias ? bias[col] : 0.0f;
    #pragma unroll
    for (int r = 0; r < 8; ++r) {
      float v = accs[t][r] + bv;
      if (relu) v = v > 0.0f ? v : 0.0f;
      C[(size_t)(rbase + r) * Nd + col] = v;
    }
  }
}

// ---------------------------------------------------------------
// Fused per-(graph,head,16-row-block) attention.
// wave: scores(16x256) -> LDS, softmax rows, O = P @ V(256x32).
// ---------------------------------------------------------------
__global__ void attn_kernel(const float* __restrict__ Q,
                            const float* __restrict__ K,
                            const float* __restrict__ V,
                            float* __restrict__ O) {
  __shared__ float smem[4 * 16 * NPG];       // 64 KB: 4 waves x (16x256)
  const int lane = threadIdx.x & 31;
  const int wid  = threadIdx.x >> 5;
  const int gw   = blockIdx.x * 4 + wid;     // [0, B*HEADS*16)
  const int rb   = gw & 15;
  const int head = (gw >> 4) & 7;
  const int b    = gw >> 7;
  float* sm = smem + wid * (16 * NPG);

  const int base = b * NPG;
  const int ml   = lane & 15;
  const int koff = (lane >> 4) << 1;
  const int rb2  = (lane >> 4) << 3;
  const int hcol = head * HDIM;

  // ---- scores: S = Qblk(16x32) @ K^T(32x256) ----
  const float* qrow = Q + (size_t)(base + rb * 16 + ml) * HID + hcol + koff;
  for (int nt = 0; nt < 16; ++nt) {
    v8f acc = {};
    const float* krow = K + (size_t)(base + nt * 16 + ml) * HID + hcol + koff;
    #pragma unroll
    for (int k0 = 0; k0 < HDIM; k0 += 4) {
      v2f a  = *(const v2f*)(qrow + k0);
      v2f bf = *(const v2f*)(krow + k0);  // (K^T)[k][n] = K[n][k]: contiguous
      acc = wmma4(a, bf, acc);
    }
    #pragma unroll
    for (int r = 0; r < 8; ++r)
      sm[(rb2 + r) * NPG + nt * 16 + ml] = acc[r];
  }
  __syncthreads();

  // ---- softmax over each of 16 rows (lanes 0-15, one row each) ----
  const float scale = 0.176776695296636881f;  // 1/sqrt(32)
  if (lane < 16) {
    float* row = sm + lane * NPG;
    float mx = -INFINITY;
    for (int j = 0; j < NPG; ++j) { float v = row[j] * scale; row[j] = v; mx = v > mx ? v : mx; }
    float sum = 0.0f;
    for (int j = 0; j < NPG; ++j) { float e = expf(row[j] - mx); row[j] = e; sum += e; }
    float inv = 1.0f / sum;
    for (int j = 0; j < NPG; ++j) row[j] *= inv;
  }
  __syncthreads();

  // ---- O = P(16x256) @ Vhead(256x32): both 16-col tiles share the P frag ----
  {
    v8f acc0 = {}, acc1 = {};
    const float* prow = sm + ml * NPG + koff;
    const float* vp0  = V + (size_t)(base + koff) * HID + hcol + ml;
    const float* vp1  = vp0 + HID;
    for (int k0 = 0; k0 < NPG; k0 += 4) {
      v2f a = *(const v2f*)(prow + k0);
      v2f b0, b1;
      b0.x = vp0[0];  b0.y = vp1[0];
      b1.x = vp0[16]; b1.y = vp1[16];
      acc0 = wmma4(a, b0, acc0);
      acc1 = wmma4(a, b1, acc1);
      vp0 += 4 * HID;
      vp1 += 4 * HID;
    }
    float* orow = O + (size_t)(base + rb * 16 + rb2) * HID + hcol + ml;
    #pragma unroll
    for (int r = 0; r < 8; ++r) {
      orow[(size_t)r * HID]      = acc0[r];
      orow[(size_t)r * HID + 16] = acc1[r];
    }
  }
}

// ---------------------------------------------------------------
// LayerNorm with residual: h = LN(h + sub) * g + b   (one wave/row)
// ---------------------------------------------------------------
__global__ void ln_residual(float* __restrict__ h, const float* __restrict__ sub,
                            const float* __restrict__ g, const float* __restrict__ b) {
  const int row  = blockIdx.x * (blockDim.x >> 5) + (threadIdx.x >> 5);
  const int lane = threadIdx.x & 31;
  float x[8];
  float s = 0.0f;
  #pragma unroll
  for (int i = 0; i < 8; ++i) {
    int c = lane + i * 32;
    x[i] = h[(size_t)row * HID + c] + sub[(size_t)row * HID + c];
    s += x[i];
  }
  for (int off = 16; off; off >>= 1) s += __shfl_xor(s, off, 32);
  float m = s * (1.0f / HID);
  float vs = 0.0f;
  #pragma unroll
  for (int i = 0; i < 8; ++i) { float d = x[i] - m; vs += d * d; }
  for (int off = 16; off; off >>= 1) vs += __shfl_xor(vs, off, 32);
  float inv = rsqrtf(vs * (1.0f / HID) + 1e-5f);
  #pragma unroll
  for (int i = 0; i < 8; ++i) {
    int c = lane + i * 32;
    h[(size_t)row * HID + c] = g[c] * (x[i] - m) * inv + b[c];
  }
}

// ---------------- graph kernels ----------------
__global__ void count_deg(const int* __restrict__ row, const int* __restrict__ col,
                          int* __restrict__ cin, int* __restrict__ cout) {
  int e = blockIdx.x * blockDim.x + threadIdx.x;
  if (e < EDG) {
    atomicAdd(&cout[row[e]], 1);
    atomicAdd(&cin[col[e]], 1);
  }
}
__global__ void max_deg(const int* __restrict__ cout, int* __restrict__ mx) {
  int i = blockIdx.x * blockDim.x + threadIdx.x;
  if (i < NN) atomicMax(mx, cout[i]);
}
__global__ void compute_dinv(const int* __restrict__ cin, float* __restrict__ dinv) {
  int i = blockIdx.x * blockDim.x + threadIdx.x;
  if (i < NN) dinv[i] = rsqrtf((float)cin[i] + 1.0f);
}
__global__ void add_pos(float* __restrict__ h, const int* __restrict__ cout,
                        const int* __restrict__ mx,
                        const float* __restrict__ pw, const float* __restrict__ pb) {
  int n = blockIdx.x, f = threadIdx.x;
  float deg  = (float)cout[n];
  float degn = deg / ((float)(*mx) + 1e-8f);
  float idx  = (float)n / (float)(NN - 1);
  float rw   = sqrtf(degn + 1e-8f);
  h[(size_t)n * HID + f] += degn * pw[f] + idx * pw[HID + f] + rw * pw[2 * HID + f] + pb[f];
}
__global__ void gcn_scatter(const float* __restrict__ xw, const int* __restrict__ row,
                            const int* __restrict__ col, const float* __restrict__ dinv,
                            float* __restrict__ agg) {
  int e = blockIdx.x, f = threadIdx.x;
  int r = row[e], c = col[e];
  float norm = dinv[r] * dinv[c];
  atomicAdd(&agg[(size_t)c * HID + f], norm * xw[(size_t)r * HID + f]);
}
__global__ void gcn_self(const float* __restrict__ xw, const float* __restrict__ dinv,
                         const float* __restrict__ bias, float* __restrict__ agg) {
  int n = blockIdx.x, f = threadIdx.x;
  float d = dinv[n];
  agg[(size_t)n * HID + f] += d * d * xw[(size_t)n * HID + f] + bias[f];
}

// ---------------- readout ----------------
__global__ void pool_kernel(const float* __restrict__ h, float* __restrict__ g) {
  int b = blockIdx.x, f = threadIdx.x;
  float s = 0.0f, mx = -INFINITY;
  for (int i = 0; i < NPG; ++i) {
    float v = h[(size_t)(b * NPG + i) * HID + f];
    s += v; mx = v > mx ? v : mx;
  }
  g[(size_t)b * 768 + f]       = s * (1.0f / NPG);
  g[(size_t)b * 768 + 256 + f] = mx;
  g[(size_t)b * 768 + 512 + f] = s;
}
__global__ void logits_kernel(const float* __restrict__ z, const float* __restrict__ w,
                              const float* __restrict__ b, float* __restrict__ out) {
  int t = threadIdx.x;
  if (t < BATCH * 2) {
    int bi = t >> 1, c = t & 1;
    float s = b[c];
    for (int k = 0; k < HID; ++k) s += z[(size_t)bi * HID + k] * w[k * 2 + c];
    out[t] = s;
  }
}

// ---------------- host-side orchestration ----------------
static inline void run_gemm(const float* A, const float* B, const float* bias,
                            float* C, int M, int Nd, int K, int relu,
                            hipStream_t stream) {
  int tiles  = (M >> 4) * (Nd >> 6);         // 16x64 strips
  int blocks = (tiles + 3) / 4;
  gemm_wmma_f32<<<blocks, 128, 0, stream>>>(A, B, bias, C, M, Nd, K, relu);
}

extern "C" void kernel_launch(void* const* d_in, const int* in_sizes, int n_in,
                              void* d_out, int out_size, void* d_ws, size_t ws_size,
                              hipStream_t stream) {
  (void)in_sizes; (void)n_in; (void)out_size; (void)ws_size;

  // inputs, setup_inputs() insertion order (params depth-first)
  const float* X    = (const float*)d_in[0];
  const float* IN_W = (const float*)d_in[1];
  const float* IN_B = (const float*)d_in[2];
  const float* POSW = (const float*)d_in[3];
  const float* POSB = (const float*)d_in[4];
  // layer l params start at 5 + 20*l:
  // 0 gcn_w 1 gcn_b 2 wq 3 bq 4 wk 5 bk 6 wv 7 bv 8 wo 9 bo
  // 10 w1 11 f_b1 12 w2 13 f_b2 14 g1 15 b1 16 g2 17 b2 18 g3 19 b3
  const float* CW1 = (const float*)d_in[85];
  const float* CB1 = (const float*)d_in[86];
  const float* CW2 = (const float*)d_in[87];
  const float* CB2 = (const float*)d_in[88];
  const float* CW3 = (const float*)d_in[89];
  const float* CB3 = (const float*)d_in[90];
  const int*   EI  = (const int*)d_in[91];    // [2, E]
  const int* erow = EI;
  const int* ecol = EI + EDG;

  // workspace carve-up
  const size_t NH = (size_t)NN * HID;           // 4,194,304 floats
  float* h    = (float*)d_ws;
  float* buf1 = h    + NH;
  float* buf2 = buf1 + NH;
  float* buf3 = buf2 + NH;
  float* buf4 = buf3 + NH;
  float* t1   = buf4 + NH;                      // N x 1024 = 4*NH floats
  float* dinv = t1 + 4 * NH;
  int*   cin  = (int*)(dinv + NN);
  int*   cout = cin + NN;
  int*   mx   = cout + NN;                      // 1 int (+pad)
  float* gp   = (float*)(mx + 64);              // [64, 768]
  float* z1   = gp + (size_t)BATCH * 768;       // [64, 512]
  float* z2   = z1 + (size_t)BATCH * 512;       // [64, 256]

  float* out_logits = (float*)d_out;
  float* out_h      = out_logits + BATCH * 2;

  // ---- degrees / norms (once) ----
  hipMemsetAsync(cin,  0, NN * sizeof(int), stream);
  hipMemsetAsync(cout, 0, NN * sizeof(int), stream);
  hipMemsetAsync(mx,   0, sizeof(int), stream);
  count_deg<<<(EDG + 255) / 256, 256, 0, stream>>>(erow, ecol, cin, cout);
  max_deg<<<(NN + 255) / 256, 256, 0, stream>>>(cout, mx);
  compute_dinv<<<(NN + 255) / 256, 256, 0, stream>>>(cin, dinv);

  // ---- input projection + positional encoding ----
  run_gemm(X, IN_W, IN_B, h, NN, HID, FIN, 0, stream);
  add_pos<<<NN, HID, 0, stream>>>(h, cout, mx, POSW, POSB);

  // ---- transformer/GCN layers ----
  for (int l = 0; l < 4; ++l) {
    const float* const* P = (const float* const*)(d_in + 5 + 20 * l);
    const float* gcn_w = P[0];  const float* gcn_b = P[1];
    const float* wq = P[2];     const float* bq = P[3];
    const float* wk = P[4];     const float* bk = P[5];
    const float* wv = P[6];     const float* bv = P[7];
    const float* wo = P[8];     const float* bo = P[9];
    const float* w1 = P[10];    const float* fb1 = P[11];
    const float* w2 = P[12];    const float* fb2 = P[13];
    const float* g1 = P[14];    const float* b1 = P[15];
    const float* g2 = P[16];    const float* b2 = P[17];
    const float* g3 = P[18];    const float* b3 = P[19];

    // GCN: xw = h @ W ; agg = scatter + self + bias ; h = LN(h + agg)
    run_gemm(h, gcn_w, nullptr, buf1, NN, HID, HID, 0, stream);
    hipMemsetAsync(buf2, 0, NH * sizeof(float), stream);
    gcn_scatter<<<EDG, HID, 0, stream>>>(buf1, erow, ecol, dinv, buf2);
    gcn_self<<<NN, HID, 0, stream>>>(buf1, dinv, gcn_b, buf2);
    ln_residual<<<NN / 4, 128, 0, stream>>>(h, buf2, g1, b1);

    // MHA: q/k/v, fused per-(graph,head) attention, out projection
    run_gemm(h, wq, bq, buf1, NN, HID, HID, 0, stream);
    run_gemm(h, wk, bk, buf2, NN, HID, HID, 0, stream);
    run_gemm(h, wv, bv, buf3, NN, HID, HID, 0, stream);
    attn_kernel<<<(BATCH * HEADS * 16) / 4, 128, 0, stream>>>(buf1, buf2, buf3, buf4);
    run_gemm(buf4, wo, bo, buf1, NN, HID, HID, 0, stream);
    ln_residual<<<NN / 4, 128, 0, stream>>>(h, buf1, g2, b2);

    // FFN
    run_gemm(h, w1, fb1, t1, NN, FFNH, HID, 1, stream);
    run_gemm(t1, w2, fb2, buf1, NN, HID, FFNH, 0, stream);
    ln_residual<<<NN / 4, 128, 0, stream>>>(h, buf1, g3, b3);
  }

  // ---- outputs: h, pooled readout, classifier ----
  hipMemcpyAsync(out_h, h, NH * sizeof(float), hipMemcpyDeviceToDevice, stream);
  pool_kernel<<<BATCH, HID, 0, stream>>>(h, gp);
  run_gemm(gp, CW1, CB1, z1, BATCH, 512, 768, 1, stream);
  run_gemm(z1, CW2, CB2, z2, BATCH, 256, 512, 1, stream);
  logits_kernel<<<1, 128, 0, stream>>>(z2, CW3, CB3, out_logits);
}